// Switch_64776696758817
// MI455X (gfx1250) — compile-verified
//
#include <hip/hip_runtime.h>
#include <hip/hip_bf16.h>
#include <stdint.h>

// diag(triggers * mask) for N=8192: 256 MB of zeros + 8192 diagonal values.
// Pure store-bandwidth problem -> stream zeros out of LDS with CDNA5 async
// LDS->global b128 stores (512 B per wave-instruction, tracked on ASYNCcnt).

#define NDIM 8192
#define THREADS 256
// 8192 floats per row = 2048 x 16B chunks; 2048 / 256 threads = 8 chunks/thread
#define CHUNKS_PER_ROW (NDIM / 4)
#define ITERS (CHUNKS_PER_ROW / THREADS)

__global__ __launch_bounds__(THREADS) void diag_fill_async(
    const float* __restrict__ triggers,
    const int*   __restrict__ vcsw_mask,
    float*       __restrict__ out)
{
    // 16B-aligned LDS sources for the async b128 stores.
    __shared__ __align__(16) float zero_chunk[4];
    __shared__ __align__(16) float diag_chunk[4];

    const unsigned row = blockIdx.x;
    const unsigned tid = threadIdx.x;

    if (tid == 0) {
        float d = triggers[row] * (float)vcsw_mask[row];
        zero_chunk[0] = 0.0f; zero_chunk[1] = 0.0f;
        zero_chunk[2] = 0.0f; zero_chunk[3] = 0.0f;
        diag_chunk[0] = 0.0f; diag_chunk[1] = 0.0f;
        diag_chunk[2] = 0.0f; diag_chunk[3] = 0.0f;
        diag_chunk[row & 3u] = d;
    }
    __syncthreads();  // DScnt barrier: LDS init visible before async engine reads

    // LDS flat-pointer low 32 bits == wave-relative LDS byte offset (shared
    // aperture lives entirely in addr[63:32] on gfx1250).
    const unsigned zoff = (unsigned)(uintptr_t)&zero_chunk[0];
    const unsigned doff = (unsigned)(uintptr_t)&diag_chunk[0];

    float* row_ptr = out + (size_t)row * NDIM;
    const unsigned diag_chunk_idx = row >> 2;  // 16B chunk containing column `row`

#pragma unroll
    for (int j = 0; j < ITERS; ++j) {
        const unsigned chunk = (unsigned)j * THREADS + tid;       // 0..2047
        const uint64_t gaddr = (uint64_t)(uintptr_t)row_ptr + (uint64_t)chunk * 16u;
        const unsigned src   = (chunk == diag_chunk_idx) ? doff : zoff;
        // Wave-wide: each lane stores 16B from the LDS address in src to the global address in gaddr.
        // Non-temporal: 256 MB output > 192 MB L2, don't pin lines.
        asm volatile("global_store_async_from_lds_b128 %0, %1, off th:TH_STORE_NT"
                     :
                     : "v"(gaddr), "v"(src)
                     : "memory");
    }

    // Drain ASYNCcnt before wave exit (S_ENDPGM would also imply wait-idle).
    asm volatile("s_wait_asynccnt 0" ::: "memory");
}

extern "C" void kernel_launch(void* const* d_in, const int* in_sizes, int n_in,
                              void* d_out, int out_size, void* d_ws, size_t ws_size,
                              hipStream_t stream) {
    (void)in_sizes; (void)n_in; (void)out_size; (void)d_ws; (void)ws_size;
    const float* triggers  = (const float*)d_in[0];
    const int*   vcsw_mask = (const int*)d_in[1];
    float*       out       = (float*)d_out;

    diag_fill_async<<<dim3(NDIM), dim3(THREADS), 0, stream>>>(triggers, vcsw_mask, out);
}